// SVMClassifier_78606491451805
// MI455X (gfx1250) — compile-verified
//
#include <hip/hip_runtime.h>
#include <math.h>

// MI455X / gfx1250, wave32. Matrix log via Chebyshev-Clenshaw (matmul-only),
// GEMMs on V_WMMA_F32_16X16X4_F32. All big buffers L2-resident (192MB L2).
// ws requirement: (3*128*65536 + 11*65536 + 10*65536 + 64) floats ~= 106 MB.

typedef float v2f __attribute__((ext_vector_type(2)));
typedef float v8f __attribute__((ext_vector_type(8)));
typedef int   v4i __attribute__((ext_vector_type(4)));
typedef __attribute__((address_space(3))) v4i lds_v4i;

#define HH   256
#define HH2  65536
#define NB   128
#define NC   10

#define APITCH 36   // A LDS pitch (words): 16B-aligned rows, conflict-free frags

// ---------------- utility kernels ----------------

__global__ void k_zero(float* __restrict__ p, long n) {
  long i = (long)blockIdx.x * blockDim.x + threadIdx.x;
  long stride = (long)gridDim.x * blockDim.x;
  for (; i < n; i += stride) p[i] = 0.0f;
}

// Wsym[c,i,j] = 0.5*(W[c,i,j] + W[c,j,i])
__global__ void k_wsym(const float* __restrict__ W, float* __restrict__ wsym) {
  int g = blockIdx.x * 256 + threadIdx.x;            // < NC*HH2
  int c = g >> 16, e = g & 65535;
  int i = e >> 8, j = e & 255;
  wsym[g] = 0.5f * (W[(c << 16) + (i << 8) + j] + W[(c << 16) + (j << 8) + i]);
}

// center = mean_b feats[b]
__global__ void k_center(const float* __restrict__ feats, float* __restrict__ center) {
  int i = blockIdx.x * 256 + threadIdx.x;            // < HH2
  float s = 0.0f;
  for (int b = 0; b < NB; ++b) s += feats[(size_t)b * HH2 + i];
  center[i] = s * (1.0f / NB);
}

// out[b,c] = <Wsym[c], feats[b]>   (feats symmetric => equals einsum cij,bji)
__global__ void k_output(const float* __restrict__ wsym, const float* __restrict__ feats,
                         float* __restrict__ out) {
  int c = blockIdx.x, b = blockIdx.y;
  const float* wp = wsym + (size_t)c * HH2;
  const float* fp = feats + (size_t)b * HH2;
  float s = 0.0f;
  for (int e = threadIdx.x; e < HH2; e += 256) s += wp[e] * fp[e];
  __shared__ float red[256];
  red[threadIdx.x] = s; __syncthreads();
  for (int o = 128; o > 0; o >>= 1) {
    if (threadIdx.x < o) red[threadIdx.x] += red[threadIdx.x + o];
    __syncthreads();
  }
  if (threadIdx.x == 0) out[b * NC + c] = red[0];
}

// gtmp[c] = sum_ij P[c,i,j]*P[c,j,i]
__global__ void k_g(const float* __restrict__ P, float* __restrict__ gtmp) {
  int c = blockIdx.x;
  const float* p = P + (size_t)c * HH2;
  float s = 0.0f;
  for (int e = threadIdx.x; e < HH2; e += 256) {
    int i = e >> 8, j = e & 255;
    s += p[e] * p[(j << 8) + i];
  }
  __shared__ float red[256];
  red[threadIdx.x] = s; __syncthreads();
  for (int o = 128; o > 0; o >>= 1) {
    if (threadIdx.x < o) red[threadIdx.x] += red[threadIdx.x + o];
    __syncthreads();
  }
  if (threadIdx.x == 0) gtmp[c] = red[0];
}

__global__ void k_final(const float* __restrict__ gtmp, float* __restrict__ out1) {
  if (threadIdx.x == 0) {
    float s = 0.0f;
    for (int c = 0; c < NC; ++c) s += gtmp[c];
    out1[0] = s * (1.0f / NC);
  }
}

// ---------------- fused 256x256 batched WMMA-f32 GEMM ----------------
// D[bz] = alpha * A[bz]*B[bz] + b1*P1[bz] + b2*P2[bz] + diagv*I
// grid: dim3(2,4,nbatch), block: 128 threads (4 waves).
// Block tile 128x64; each wave: 2 M-tiles x 4 N-tiles (8 x v8f acc).
// A panel staged via async global->LDS (B128, ASYNCcnt); B packed as
// K-pair float2 with pitch 80 so each B fragment is one ds_load_b64.
__global__ __launch_bounds__(128) void k_gemm256(
    const float* __restrict__ A,  long sA,
    const float* __restrict__ Bm, long sB,
    const float* __restrict__ P1, const float* __restrict__ P2, long sP,
    float* __restrict__ D, long sD,
    float alpha, float b1, float b2, float diagv)
{
  __shared__ float  As[128 * APITCH];  // 18432 B
  __shared__ float2 Bs2[16 * 80];      // 10240 B

  const int tid  = threadIdx.x;
  const int l    = tid & 31;
  const int w    = tid >> 5;          // wave 0..3 -> 32-row M strip
  const int half = l >> 4;            // WMMA lane half
  const int mn   = l & 15;
  const int bz   = blockIdx.z;
  const int tmBase = blockIdx.x * 128;
  const int tnBase = blockIdx.y * 64;

  const float* Ag = A  + (size_t)bz * sA;
  const float* Bg = Bm + (size_t)bz * sB;

  v8f acc[2][4];
  #pragma unroll
  for (int mt = 0; mt < 2; ++mt)
    #pragma unroll
    for (int nt = 0; nt < 4; ++nt) acc[mt][nt] = (v8f){};

  const int w32 = w << 5;

  for (int kc = 0; kc < HH; kc += 32) {
    // prefetch next K panel into caches (global_prefetch_b8 path)
    if (kc + 32 < HH) {
      __builtin_prefetch(Ag + (size_t)(tmBase + tid) * HH + kc + 32, 0, 3);
      if (tid < 64)
        __builtin_prefetch(Bg + (size_t)(kc + 32 + (tid >> 1)) * HH + tnBase + (tid & 1) * 32, 0, 3);
    }

    // ---- 1) issue B panel global loads (registers) ----
    float4 bv[4];
    #pragma unroll
    for (int s = 0; s < 4; ++s) {
      int slot = tid + s * 128;
      int row = slot >> 4, c4 = slot & 15;
      bv[s] = *(const float4*)(Bg + (size_t)(kc + row) * HH + tnBase + (c4 << 2));
    }

    // ---- 2) issue async A panel global->LDS (1024 B128 slots) ----
#if defined(__has_builtin) && __has_builtin(__builtin_amdgcn_global_load_async_to_lds_b128)
    #pragma unroll
    for (int s = 0; s < 8; ++s) {
      int slot = tid + s * 128;
      int row = slot >> 3, c4 = slot & 7;
      __builtin_amdgcn_global_load_async_to_lds_b128(
          (v4i*)(Ag + (size_t)(tmBase + row) * HH + kc + (c4 << 2)),
          (lds_v4i*)&As[row * APITCH + (c4 << 2)],
          0, 0);
    }
#else
    #pragma unroll
    for (int s = 0; s < 8; ++s) {
      int slot = tid + s * 128;
      int row = slot >> 3, c4 = slot & 7;
      const float4 av = *(const float4*)(Ag + (size_t)(tmBase + row) * HH + kc + (c4 << 2));
      *(float4*)&As[row * APITCH + (c4 << 2)] = av;  // 16B-aligned (pitch 36)
    }
#endif

    // ---- 3) scatter B into LDS, K-pair packed: Bs2[k>>1][n]={B[k][n],B[k+1][n]}
    //         (overlaps with the in-flight async A DMA)
    #pragma unroll
    for (int s = 0; s < 4; ++s) {
      int slot = tid + s * 128;
      int row = slot >> 4, c4 = slot & 15;
      float* bsf = (float*)&Bs2[(row >> 1) * 80 + (c4 << 2)];
      const int comp = row & 1;
      bsf[0 + comp] = bv[s].x;
      bsf[2 + comp] = bv[s].y;
      bsf[4 + comp] = bv[s].z;
      bsf[6 + comp] = bv[s].w;
    }

    // ---- 4) fence async DMA, then block barrier ----
#if defined(__has_builtin) && __has_builtin(__builtin_amdgcn_global_load_async_to_lds_b128)
    asm volatile("s_wait_asynccnt 0x0" ::: "memory");
#endif
    __syncthreads();

    #pragma unroll
    for (int kk = 0; kk < 8; ++kk) {
      const int kb = (kk << 2) + (half << 1);   // K base for this lane half
      const int kp = (kk << 1) + half;          // packed-pair row
      v2f a0 = *(const v2f*)&As[(w32 + mn) * APITCH + kb];
      v2f a1 = *(const v2f*)&As[(w32 + 16 + mn) * APITCH + kb];
      const float2* brow = &Bs2[kp * 80 + mn];
      #pragma unroll
      for (int nt = 0; nt < 4; ++nt) {
        v2f b = *(const v2f*)&brow[nt << 4];    // single ds_load_b64
        acc[0][nt] = __builtin_amdgcn_wmma_f32_16x16x4_f32(
            false, a0, false, b, (short)0, acc[0][nt], false, false);
        acc[1][nt] = __builtin_amdgcn_wmma_f32_16x16x4_f32(
            false, a1, false, b, (short)0, acc[1][nt], false, false);
      }
    }
    __syncthreads();
  }

  // epilogue
  const float* p1 = P1 + (size_t)bz * sP;
  const float* p2 = P2 + (size_t)bz * sP;
  float* d = D + (size_t)bz * sD;
  #pragma unroll
  for (int mt = 0; mt < 2; ++mt) {
    #pragma unroll
    for (int nt = 0; nt < 4; ++nt) {
      #pragma unroll
      for (int r = 0; r < 8; ++r) {
        int row = tmBase + w32 + (mt << 4) + (half << 3) + r;  // M = r + 8*half
        int col = tnBase + (nt << 4) + mn;
        int idx = (row << 8) + col;
        float v = alpha * acc[mt][nt][r] + b1 * p1[idx] + b2 * p2[idx];
        if (row == col) v += diagv;
        d[idx] = v;
      }
    }
  }
}

// ---------------- host ----------------

extern "C" void kernel_launch(void* const* d_in, const int* in_sizes, int n_in,
                              void* d_out, int out_size, void* d_ws, size_t ws_size,
                              hipStream_t stream) {
  const float* X = (const float*)d_in[0];   // [128,256,256] SPD, eig in [1, ~5.1]
  const float* W = (const float*)d_in[1];   // [10,256,256]
  float* out = (float*)d_out;               // 1281 floats
  float* ws  = (float*)d_ws;

  const long S = (long)NB * HH2;            // per-buffer floats
  float* buf[3] = { ws, ws + S, ws + 2 * S };
  float* wsym   = ws + 3 * S;               // NC*HH2
  float* center = wsym + (long)NC * HH2;    // HH2
  float* Pm     = center + HH2;             // NC*HH2
  float* gtmp   = Pm + (long)NC * HH2;      // NC

  // Chebyshev coefficients of log(x) on [1, 5.5] (closed form via
  // sum T_n(t) z^n/n = -0.5*ln(1-2tz+z^2)); degree 20, tail < 1e-9.
  const int N = 20;
  const double lo = 1.0, hi = 5.5;
  const double cc = 0.5 * (lo + hi), hh = 0.5 * (hi - lo);
  const double mr = hh / cc;
  const double zz = -mr / (1.0 + sqrt(1.0 - mr * mr));
  double a[N + 1];
  a[0] = log(cc) - log(1.0 + zz * zz);
  double zp = 1.0;
  for (int n = 1; n <= N; ++n) { zp *= zz; a[n] = -2.0 * zp / n; }

  // init: Clenshaw b_{N+1} = b_{N+2} = 0
  k_zero<<<4096, 256, 0, stream>>>(buf[0], 2 * S);
  k_wsym<<<(NC * HH2) / 256, 256, 0, stream>>>(W, wsym);

  // Clenshaw: b_k = a_k I + (2/h) X b_{k+1} - (2c/h) b_{k+1} - b_{k+2}
  int ip2 = 0, ip1 = 1;
  const float al  = (float)(2.0 / hh);
  const float bb1 = (float)(-2.0 * cc / hh);
  dim3 gg(2, 4, NB);
  for (int k = N; k >= 1; --k) {
    int in_ = 3 - ip1 - ip2;
    k_gemm256<<<gg, 128, 0, stream>>>(X, HH2, buf[ip1], HH2,
                                      buf[ip1], buf[ip2], HH2,
                                      buf[in_], HH2,
                                      al, bb1, -1.0f, (float)a[k]);
    ip2 = ip1; ip1 = in_;
  }
  // feats = logm(X) = a_0 I + (1/h) X b_1 - (c/h) b_1 - b_2
  int ifeat = 3 - ip1 - ip2;
  k_gemm256<<<gg, 128, 0, stream>>>(X, HH2, buf[ip1], HH2,
                                    buf[ip1], buf[ip2], HH2,
                                    buf[ifeat], HH2,
                                    (float)(1.0 / hh), (float)(-cc / hh), -1.0f, (float)a[0]);
  float* feats = buf[ifeat];

  k_center<<<HH2 / 256, 256, 0, stream>>>(feats, center);
  k_output<<<dim3(NC, NB), 256, 0, stream>>>(wsym, feats, out);
  // P[c] = Wsym[c] @ center
  k_gemm256<<<dim3(2, 4, NC), 128, 0, stream>>>(wsym, HH2, center, 0,
                                                center, center, 0,
                                                Pm, HH2,
                                                1.0f, 0.0f, 0.0f, 0.0f);
  k_g<<<NC, 256, 0, stream>>>(Pm, gtmp);
  k_final<<<1, 32, 0, stream>>>(gtmp, out + NB * NC);
}